// VolumeConvolutionF_12146167513334
// MI455X (gfx1250) — compile-verified
//
#include <hip/hip_runtime.h>
#include <math.h>

// ---------------------------------------------------------------------------
// Batched 3D circular cross-correlation via DFT-as-GEMM on CDNA5 WMMA.
//   out = IFFT3( conj(FFT3(v1)) * FFT3(v2) )
// Each 1D length-128 DFT along an axis is a 128x128 f32 matmul applied to
// B*S^2 = 131072 vectors -> batched complex GEMM on V_WMMA_F32_16X16X4_F32.
// Per block: one 16-vector tile staged through LDS, 8 waves cover all 128
// output frequencies (kTile 0..7) from the shared tile.
// ---------------------------------------------------------------------------

typedef __attribute__((ext_vector_type(2))) float v2f;
typedef __attribute__((ext_vector_type(8))) float v8f;

#define SDIM   128
#define BATCH  8
#define NTOT   (BATCH * SDIM * SDIM * SDIM)   // 16,777,216 elements per volume
#define NVEC   (BATCH * SDIM * SDIM)          // 131,072 vectors per axis pass
#define WN     (SDIM * SDIM)                  // 16,384 elements per DFT matrix

// LDS tile: X[n][l], n = 0..15 vectors, l = 0..127 points, re plane then im.
// Pitch 132 floats: B-fragment ds_load_b64 hits bank (4*lid + k) % 64 ->
// all 64 banks exactly once per wave32 access, conflict-free; rows stay
// 16B-aligned (132*4 = 528 = 33*16) for float4 stores.
#define LPITCH 132
#define XOFF   (16 * LPITCH)                  // start of imaginary plane

// ---------------------------------------------------------------------------
// Twiddle matrices. Forward: W[k][l] = exp(-2*pi*i*k*l/S).
// Inverse:  W[k][l] = exp(+2*pi*i*k*l/S) / S   (1/S per axis -> 1/S^3 total).
// ---------------------------------------------------------------------------
__global__ void gen_dft_mats(float* __restrict__ wf_r, float* __restrict__ wf_i,
                             float* __restrict__ wb_r, float* __restrict__ wb_i) {
  int idx = blockIdx.x * blockDim.x + threadIdx.x;
  if (idx >= WN) return;
  int k = idx >> 7;
  int l = idx & 127;
  double ang = -2.0 * 3.14159265358979323846 * (double)(k * l) / (double)SDIM;
  wf_r[idx] = (float)cos(ang);
  wf_i[idx] = (float)sin(ang);
  wb_r[idx] = (float)(cos(-ang) / (double)SDIM);
  wb_i[idx] = (float)(sin(-ang) / (double)SDIM);
}

// real volume -> (re, im=0) planes
__global__ void pack_complex(const float* __restrict__ in,
                             float* __restrict__ outr, float* __restrict__ outi) {
  int i = blockIdx.x * blockDim.x + threadIdx.x;
  outr[i] = in[i];
  outi[i] = 0.0f;
}

// pointwise conj(F1) * F2
__global__ void conj_mul(const float* __restrict__ f1r, const float* __restrict__ f1i,
                         const float* __restrict__ f2r, const float* __restrict__ f2i,
                         float* __restrict__ outr, float* __restrict__ outi) {
  int i = blockIdx.x * blockDim.x + threadIdx.x;
  float ar = f1r[i], ai = f1i[i];
  float br = f2r[i], bi = f2i[i];
  outr[i] = ar * br + ai * bi;   // Re(conj(a)*b)
  outi[i] = ar * bi - ai * br;   // Im(conj(a)*b)
}

// ---------------------------------------------------------------------------
// One DFT pass along an axis with element stride D = 1<<shift.
// Vector v (the B*S^2 non-axis coordinates) has base address
//   base(v) = (v / D) * (D * S) + (v % D),  element l at base(v) + l*D.
// Block tile: 16 vectors (vTile = blockIdx.x) staged into LDS once, then
// wave w (= threadIdx.x>>5) computes output frequencies kTile = w.
// Complex GEMM via 4 real f32 WMMA accumulators:
//   Yr = Wr*Xr - Wi*Xi ; Yi = Wr*Xi + Wi*Xr
// Operand layouts per ISA 7.12.2 (f32, wave32):
//   A 16x4 : lanes 0-15 -> K={0,1} in VGPR{0,1}; lanes 16-31 -> K={2,3}
//   B 4x16 : mirrored (lane = column N, VGPR pair = K within half)
//   C/D    : VGPR i -> M = i + 8*(lane>=16), N = lane&15
// ---------------------------------------------------------------------------
__global__ __launch_bounds__(256)
void dft_axis_wmma(const float* __restrict__ xr, const float* __restrict__ xi,
                   float* __restrict__ yr, float* __restrict__ yi,
                   const float* __restrict__ wr, const float* __restrict__ wi,
                   int shift) {
  __shared__ float Xs[2 * 16 * LPITCH];   // 16.9 KB: re plane then im plane

  const int tid = threadIdx.x;
  const int v0  = blockIdx.x * 16;        // first vector of this block's tile
  const int D   = 1 << shift;

  // ---- cooperative, coalesced global -> LDS staging of the X tile ----
  if (shift == 0) {
    // D = 1: vector (v0+n) is 128 contiguous floats at (v0+n)*128.
    // 16 threads per vector, 8 contiguous floats each -> 512B/row coalesced.
    const int n  = tid >> 4;
    const int l0 = (tid & 15) * 8;
    const float4* sr = (const float4*)(xr + (size_t)(v0 + n) * SDIM + l0);
    const float4* si = (const float4*)(xi + (size_t)(v0 + n) * SDIM + l0);
    float4 r0 = sr[0], r1 = sr[1];
    float4 i0 = si[0], i1 = si[1];
    float* dr = &Xs[n * LPITCH + l0];
    float* di = &Xs[XOFF + n * LPITCH + l0];
    *(float4*)(dr)     = r0;
    *(float4*)(dr + 4) = r1;
    *(float4*)(di)     = i0;
    *(float4*)(di + 4) = i1;
  } else {
    // D >= 128: for fixed l, the 16 vectors are 16 consecutive floats.
    // v0 is a multiple of 16 and D >= 128, so base(v0+n) = base(v0) + n.
    const int base0 = ((v0 >> shift) << (shift + 7)) + (v0 & (D - 1));
    const int n    = tid & 15;
    const int lrow = tid >> 4;            // 0..15
#pragma unroll
    for (int it = 0; it < 8; ++it) {
      const int l = it * 16 + lrow;
      const int g = base0 + n + l * D;
      Xs[n * LPITCH + l]        = xr[g];
      Xs[XOFF + n * LPITCH + l] = xi[g];
    }
  }
  __syncthreads();

  // ---- per-wave WMMA complex GEMM on the shared tile ----
  const int lane  = tid & 31;
  const int kTile = tid >> 5;             // 0..7: which 16 frequencies
  const int lid   = lane & 15;
  const int hi    = lane >> 4;

  // A operand: twiddle rows (global, L2/WGP$ resident, reused 8192x).
  const int wrow = kTile * 16 + lid;
  const float* wrp = wr + wrow * SDIM + 2 * hi;
  const float* wip = wi + wrow * SDIM + 2 * hi;

  // B operand: column n = lid of the LDS tile.
  const float* xsr = &Xs[lid * LPITCH + 2 * hi];
  const float* xsi = &Xs[XOFF + lid * LPITCH + 2 * hi];

  v8f acc_rr = {}, acc_ii = {}, acc_ri = {}, acc_ir = {};

  for (int kb = 0; kb < SDIM; kb += 4) {
    v2f ar = *(const v2f*)(wrp);
    v2f ai = *(const v2f*)(wip);
    v2f br = *(const v2f*)(xsr + kb);     // ds_load_b64, conflict-free
    v2f bi = *(const v2f*)(xsi + kb);
    wrp += 4; wip += 4;

    acc_rr = __builtin_amdgcn_wmma_f32_16x16x4_f32(false, ar, false, br,
                                                   (short)0, acc_rr, false, false);
    acc_ii = __builtin_amdgcn_wmma_f32_16x16x4_f32(false, ai, false, bi,
                                                   (short)0, acc_ii, false, false);
    acc_ri = __builtin_amdgcn_wmma_f32_16x16x4_f32(false, ar, false, bi,
                                                   (short)0, acc_ri, false, false);
    acc_ir = __builtin_amdgcn_wmma_f32_16x16x4_f32(false, ai, false, br,
                                                   (short)0, acc_ir, false, false);
  }

  // Writeback: combine the four real accumulators into complex output.
  const int v    = v0 + lid;
  const int base = ((v >> shift) << (shift + 7)) + (v & (D - 1));
#pragma unroll
  for (int i = 0; i < 8; ++i) {
    const int krow = kTile * 16 + i + 8 * hi;
    const int addr = base + krow * D;
    yr[addr] = acc_rr[i] - acc_ii[i];
    yi[addr] = acc_ri[i] + acc_ir[i];
  }
}

// ---------------------------------------------------------------------------
extern "C" void kernel_launch(void* const* d_in, const int* in_sizes, int n_in,
                              void* d_out, int out_size, void* d_ws, size_t ws_size,
                              hipStream_t stream) {
  (void)in_sizes; (void)n_in; (void)out_size; (void)ws_size;

  const float* v1 = (const float*)d_in[0];
  const float* v2 = (const float*)d_in[1];
  float* out = (float*)d_out;
  float* ws  = (float*)d_ws;

  // Workspace layout (floats):
  float* wf_r = ws;             // forward twiddles, re
  float* wf_i = ws + 1 * WN;    // forward twiddles, im
  float* wb_r = ws + 2 * WN;    // inverse twiddles (incl. 1/S), re
  float* wb_i = ws + 3 * WN;    // inverse twiddles (incl. 1/S), im
  float* P0 = ws + 4 * WN;
  float* P1 = P0 + (size_t)NTOT;
  float* P2 = P1 + (size_t)NTOT;
  float* P3 = P2 + (size_t)NTOT;
  float* P4 = P3 + (size_t)NTOT;
  float* P5 = P4 + (size_t)NTOT;

  const dim3 blk(256);

  gen_dft_mats<<<WN / 256, blk, 0, stream>>>(wf_r, wf_i, wb_r, wb_i);

  const int ewBlocks  = NTOT / 256;   // elementwise kernels
  const int dftBlocks = NVEC / 16;    // one block per 16-vector tile (8192)

  // ---- forward FFT3(v1) -> (P2, P3) ----
  pack_complex<<<ewBlocks, blk, 0, stream>>>(v1, P0, P1);
  dft_axis_wmma<<<dftBlocks, blk, 0, stream>>>(P0, P1, P2, P3, wf_r, wf_i, 0);
  dft_axis_wmma<<<dftBlocks, blk, 0, stream>>>(P2, P3, P0, P1, wf_r, wf_i, 7);
  dft_axis_wmma<<<dftBlocks, blk, 0, stream>>>(P0, P1, P2, P3, wf_r, wf_i, 14);

  // ---- forward FFT3(v2) -> (P4, P5) ----
  pack_complex<<<ewBlocks, blk, 0, stream>>>(v2, P0, P1);
  dft_axis_wmma<<<dftBlocks, blk, 0, stream>>>(P0, P1, P4, P5, wf_r, wf_i, 0);
  dft_axis_wmma<<<dftBlocks, blk, 0, stream>>>(P4, P5, P0, P1, wf_r, wf_i, 7);
  dft_axis_wmma<<<dftBlocks, blk, 0, stream>>>(P0, P1, P4, P5, wf_r, wf_i, 14);

  // ---- pointwise conj(F1)*F2 -> (P0, P1) ----
  conj_mul<<<ewBlocks, blk, 0, stream>>>(P2, P3, P4, P5, P0, P1);

  // ---- inverse FFT3 -> real part straight into d_out ----
  dft_axis_wmma<<<dftBlocks, blk, 0, stream>>>(P0, P1, P2, P3, wb_r, wb_i, 0);
  dft_axis_wmma<<<dftBlocks, blk, 0, stream>>>(P2, P3, P0, P1, wb_r, wb_i, 7);
  dft_axis_wmma<<<dftBlocks, blk, 0, stream>>>(P0, P1, out, P3, wb_r, wb_i, 14);
}